// ConvertToVector_15968688406880
// MI455X (gfx1250) — compile-verified
//
#include <hip/hip_runtime.h>
#include <hip/hip_bf16.h>
#include <stdint.h>

// im2col patch extraction for x=(16,64,128,128) f32 -> out=(16,128,128,64,3,3).
// Zero-FLOP permutation: bound by HBM stores (576 MiB out + 64 MiB in ~ 29us
// floor at 23.3 TB/s). Per-block 32-pixel row tile:
//   * stage 3x64x(32+2) input halo into LDS with CDNA5 async-to-LDS loads
//     (b128 for the aligned interior, b32 for the 2 halo columns; ASYNCcnt)
//   * emit one contiguous 73,728-B output region with 16B-aligned
//     non-temporal b128 stores (input stays L2-resident, output streams).

#define BATCH 16
#define CH 64
#define HT 128
#define WD 128
#define TW 32            // pixels per block tile (one row segment)
#define PITCH 40         // cols 4..35 = interior (16B aligned), 3 & 36 = halo
#define NTHREADS 256

#define OUT_FLOATS (TW * CH * 9)   // 18432 output floats per tile

typedef __attribute__((address_space(1))) int g_int;
typedef __attribute__((address_space(3))) int l_int;
typedef float v4f __attribute__((ext_vector_type(4)));
typedef int v4i __attribute__((ext_vector_type(4)));
typedef __attribute__((address_space(1))) v4i g_v4i;
typedef __attribute__((address_space(3))) v4i l_v4i;

__device__ __forceinline__ void wait_async_all() {
#if __has_builtin(__builtin_amdgcn_s_wait_asynccnt)
  __builtin_amdgcn_s_wait_asynccnt(0);
#else
  asm volatile("s_wait_asynccnt 0" ::: "memory");
#endif
}

__global__ __launch_bounds__(NTHREADS) void im2col_tile_kernel(
    const float* __restrict__ x, float* __restrict__ out) {
  __shared__ float lds[3 * CH * PITCH];  // 30,720 bytes

  const int tile = blockIdx.x;            // B * H * (W/TW) = 8192 tiles
  const int wt = tile & 3;                // W/TW == 4
  const int h = (tile >> 2) & (HT - 1);
  const int b = tile >> 9;                // / (4*128)
  const int w0 = wt * TW;
  const int tid = threadIdx.x;

  // ---- Stage interior: 3 rows x 64 ch x 32 cols as 16B chunks (async b128) --
  // LDS col (4 + 4k + i) holds w = w0 + 4k + i  (i.e. col c -> w = w0 + c - 4).
  for (int idx = tid; idx < 3 * CH * 8; idx += NTHREADS) {  // 6 iters/thread
    const int chunk = idx & 7;
    const int t = idx >> 3;
    const int c = t & (CH - 1);
    const int di = t >> 6;  // 0..2
    const int gh = h - 1 + di;
    const int loff = (di * CH + c) * PITCH + 4 + chunk * 4;  // 16B aligned
    if ((unsigned)gh < (unsigned)HT) {
      const float* gp =
          x + (((size_t)(b * CH + c) * HT + gh) * WD + w0 + chunk * 4);
#if __has_builtin(__builtin_amdgcn_global_load_async_to_lds_b128)
      __builtin_amdgcn_global_load_async_to_lds_b128(
          (g_v4i*)gp, (l_v4i*)&lds[loff], /*offset=*/0, /*cpol=*/0);
#elif __has_builtin(__builtin_amdgcn_global_load_async_to_lds_b32)
      for (int i = 0; i < 4; ++i)
        __builtin_amdgcn_global_load_async_to_lds_b32(
            (g_int*)(gp + i), (l_int*)&lds[loff + i], 0, 0);
#else
      *(v4f*)&lds[loff] = *(const v4f*)gp;
#endif
    } else {
      *(v4f*)&lds[loff] = (v4f)0.0f;  // SAME padding (top/bottom rows)
    }
  }

  // ---- Stage halo: left col (w0-1 -> col 3), right col (w0+32 -> col 36) ----
  for (int idx = tid; idx < 3 * CH * 2; idx += NTHREADS) {
    const int side = idx & 1;
    const int t = idx >> 1;
    const int c = t & (CH - 1);
    const int di = t >> 6;
    const int gh = h - 1 + di;
    const int gw = side ? (w0 + TW) : (w0 - 1);
    const int loff = (di * CH + c) * PITCH + (side ? 36 : 3);
    if ((unsigned)gh < (unsigned)HT && (unsigned)gw < (unsigned)WD) {
      const float* gp = x + (((size_t)(b * CH + c) * HT + gh) * WD + gw);
#if __has_builtin(__builtin_amdgcn_global_load_async_to_lds_b32)
      __builtin_amdgcn_global_load_async_to_lds_b32(
          (g_int*)gp, (l_int*)&lds[loff], /*offset=*/0, /*cpol=*/0);
#else
      lds[loff] = *gp;
#endif
    } else {
      lds[loff] = 0.0f;  // SAME padding (left/right edges)
    }
  }

  wait_async_all();   // drain this wave's ASYNCcnt before signaling barrier
  __syncthreads();    // all LDS writes (ds + async) now visible

  // ---- Emit one contiguous 32*576-float output region via NT b128 stores ----
  float* outTile = out + (size_t)((b * HT + h) * WD + w0) * (CH * 9);
  for (int v = tid; v < OUT_FLOATS / 4; v += NTHREADS) {  // 18 iters/thread
    const int f = v * 4;
    v4f r;
#pragma unroll
    for (int l = 0; l < 4; ++l) {
      const int ff = f + l;
      const int pix = ff / 576;          // pixel within tile (0..31)
      const int rem = ff - pix * 576;
      const int c = rem / 9;
      const int e = rem - c * 9;         // i*3 + j
      const int di = e / 3;
      const int dj = e - di * 3;
      r[l] = lds[(di * CH + c) * PITCH + pix + dj + 3];  // col 3 == w0-1
    }
    __builtin_nontemporal_store(r, (v4f*)(outTile + f));
  }
}

extern "C" void kernel_launch(void* const* d_in, const int* in_sizes, int n_in,
                              void* d_out, int out_size, void* d_ws,
                              size_t ws_size, hipStream_t stream) {
  const float* x = (const float*)d_in[0];
  float* out = (float*)d_out;
  const int n_tiles = BATCH * HT * (WD / TW);  // 8192
  im2col_tile_kernel<<<n_tiles, NTHREADS, 0, stream>>>(x, out);
}